// Deepfluid_81638738362624
// MI455X (gfx1250) — compile-verified
//
#include <hip/hip_runtime.h>
#include <hip/hip_bf16.h>
#include <math.h>

typedef __attribute__((ext_vector_type(2))) float v2f;
typedef __attribute__((ext_vector_type(8))) float v8f;

#define RADIUS  1.8f
#define KNEIGH  16
#define KS      4

// ---------------------------------------------------------------------------
// Layer 1: C=2 is too small for the matrix pipe -> pure VALU, one thread/query.
// h1[n, 0:32]  = relu(box cconv), h1[n,32:64] = relu(dy cconv),
// h1[n,64:96]  = relu(dy_feats @ W1 + b1)
// ---------------------------------------------------------------------------
__global__ __launch_bounds__(256)
void layer1_kernel(const float* __restrict__ qpos, const float* __restrict__ dyfeat,
                   const float* __restrict__ boxpos, const float* __restrict__ boxfeat,
                   const int* __restrict__ dyidx, const int* __restrict__ boxidx,
                   const float* __restrict__ W1, const float* __restrict__ b1,
                   const float* __restrict__ Wc1, float* __restrict__ h1, int N)
{
    __shared__ float sWc1[KS * 2 * 32];
    __shared__ float sW1[2 * 32];
    __shared__ float sb1[32];
    int tid = threadIdx.x;
    if (tid < KS * 2 * 32) sWc1[tid] = Wc1[tid];
    if (tid < 64)          sW1[tid]  = W1[tid];
    if (tid < 32)          sb1[tid]  = b1[tid];
    __syncthreads();

    int n = blockIdx.x * 256 + tid;
    if (n >= N) return;

    float qx = qpos[n * 3 + 0], qy = qpos[n * 3 + 1], qz = qpos[n * 3 + 2];

    float accb[KS][2], accd[KS][2];
#pragma unroll
    for (int b = 0; b < KS; b++) { accb[b][0] = accb[b][1] = 0.f; accd[b][0] = accd[b][1] = 0.f; }

    // boundary cconv accumulation
    for (int k = 0; k < KNEIGH; k++) {
        int j = boxidx[n * KNEIGH + k];
        float dx = boxpos[j * 3 + 0] - qx;
        float dy = boxpos[j * 3 + 1] - qy;
        float dz = boxpos[j * 3 + 2] - qz;
        float d  = sqrtf(dx * dx + dy * dy + dz * dz);
        float t  = fminf(d * (1.0f / RADIUS), 1.0f) * (float)(KS - 1);
        float f0 = boxfeat[j * 2 + 0], f1 = boxfeat[j * 2 + 1];
#pragma unroll
        for (int b = 0; b < KS; b++) {
            float a = fmaxf(0.0f, 1.0f - fabsf(t - (float)b)) * (1.0f / KNEIGH);
            accb[b][0] += a * f0;
            accb[b][1] += a * f1;
        }
    }
    // fluid self cconv accumulation
    for (int k = 0; k < KNEIGH; k++) {
        int j = dyidx[n * KNEIGH + k];
        float dx = qpos[j * 3 + 0] - qx;
        float dy = qpos[j * 3 + 1] - qy;
        float dz = qpos[j * 3 + 2] - qz;
        float d  = sqrtf(dx * dx + dy * dy + dz * dz);
        float t  = fminf(d * (1.0f / RADIUS), 1.0f) * (float)(KS - 1);
        float f0 = dyfeat[j * 2 + 0], f1 = dyfeat[j * 2 + 1];
#pragma unroll
        for (int b = 0; b < KS; b++) {
            float a = fmaxf(0.0f, 1.0f - fabsf(t - (float)b)) * (1.0f / KNEIGH);
            accd[b][0] += a * f0;
            accd[b][1] += a * f1;
        }
    }

    float sf0 = dyfeat[n * 2 + 0], sf1 = dyfeat[n * 2 + 1];
    float* o = h1 + (size_t)n * 96;
    for (int d = 0; d < 32; d++) {
        float bo = 0.f, dv = 0.f;
#pragma unroll
        for (int b = 0; b < KS; b++) {
            bo += accb[b][0] * sWc1[(b * 2 + 0) * 32 + d] + accb[b][1] * sWc1[(b * 2 + 1) * 32 + d];
            dv += accd[b][0] * sWc1[(b * 2 + 0) * 32 + d] + accd[b][1] * sWc1[(b * 2 + 1) * 32 + d];
        }
        float sl = sf0 * sW1[d] + sf1 * sW1[32 + d] + sb1[d];
        o[d]      = fmaxf(bo, 0.0f);
        o[32 + d] = fmaxf(dv, 0.0f);
        o[64 + d] = fmaxf(sl, 0.0f);
    }
}

// ---------------------------------------------------------------------------
// Fused cconv + linear layer:
//   out = act(cconv(sfeat, Wc)) + sfeat_q @ W + b
// Phase 1/2: VALU gather + hat-basis accumulation into LDS acc tile [32, 4C].
// Phase 3:   V_WMMA_F32_16X16X4_F32 GEMMs  [32,4C]x[4C,D]  and  [32,C]x[C,D].
// 8 wave32 per block: 2 row-tiles x ceil(D/16) col-tiles of 16x16 outputs.
// ---------------------------------------------------------------------------
template <int C, int D, bool RELU_CC>
__global__ __launch_bounds__(256)
void cconv_layer(const float* __restrict__ qpos, const float* __restrict__ spos,
                 const float* __restrict__ sfeat, const int* __restrict__ idx,
                 const float* __restrict__ Wc, const float* __restrict__ W,
                 const float* __restrict__ bias, float* __restrict__ out, int N)
{
    constexpr int NT   = 32;           // queries per block
    constexpr int ASTR = 4 * C + 2;    // acc row stride (padded: breaks LDS bank conflicts)
    constexpr int HSTR = C + 1;        // h_in row stride
    constexpr int CT   = (D + 15) / 16;

    __shared__ float s_acc[NT * ASTR];
    __shared__ float s_hin[NT * HSTR];
    __shared__ float s_a[NT][KNEIGH][KS];
    __shared__ int   s_idx[NT][KNEIGH];

    const int tid = threadIdx.x;
    const int n0  = blockIdx.x * NT;

    // ---- phase 1: hat-basis weights (1/K folded in) + neighbor indices ----
    for (int p = tid; p < NT * KNEIGH; p += 256) {
        int nl = p / KNEIGH, k = p % KNEIGH;
        int gn = n0 + nl;
        if (gn < N) {
            int j = idx[gn * KNEIGH + k];
            s_idx[nl][k] = j;
            float dx = spos[j * 3 + 0] - qpos[gn * 3 + 0];
            float dy = spos[j * 3 + 1] - qpos[gn * 3 + 1];
            float dz = spos[j * 3 + 2] - qpos[gn * 3 + 2];
            float d  = sqrtf(dx * dx + dy * dy + dz * dz);
            float t  = fminf(d * (1.0f / RADIUS), 1.0f) * (float)(KS - 1);
#pragma unroll
            for (int b = 0; b < KS; b++)
                s_a[nl][k][b] = fmaxf(0.0f, 1.0f - fabsf(t - (float)b)) * (1.0f / KNEIGH);
        } else {
            s_idx[nl][k] = 0;
#pragma unroll
            for (int b = 0; b < KS; b++) s_a[nl][k][b] = 0.0f;
        }
    }
    // query self-features for the fused linear term
    for (int p = tid; p < NT * C; p += 256) {
        int nl = p / C, c = p % C;
        int gn = n0 + nl;
        s_hin[nl * HSTR + c] = (gn < N) ? sfeat[(size_t)gn * C + c] : 0.0f;
    }
    __syncthreads();

    // ---- phase 2: gather + accumulate acc[n][b*C+c] (gathers hit 192MB L2) ----
    for (int p = tid; p < NT * C; p += 256) {
        int nl = p / C, c = p % C;
        float a0 = 0.f, a1 = 0.f, a2 = 0.f, a3 = 0.f;
#pragma unroll
        for (int k = 0; k < KNEIGH; k++) {
            float f = sfeat[(size_t)s_idx[nl][k] * C + c];
            a0 += s_a[nl][k][0] * f;
            a1 += s_a[nl][k][1] * f;
            a2 += s_a[nl][k][2] * f;
            a3 += s_a[nl][k][3] * f;
        }
        s_acc[nl * ASTR + 0 * C + c] = a0;
        s_acc[nl * ASTR + 1 * C + c] = a1;
        s_acc[nl * ASTR + 2 * C + c] = a2;
        s_acc[nl * ASTR + 3 * C + c] = a3;
    }
    __syncthreads();

    // ---- phase 3: f32 WMMA GEMMs (wave-uniform branch -> EXEC all 1s) ----
    const int w    = tid >> 5;
    const int lane = tid & 31;
    if (w < 2 * CT) {
        const int rt   = w / CT;        // row tile (0/1)
        const int ct   = w % CT;        // col tile
        const int half = lane >> 4;     // A: K pair select; C/D: M+8 select
        const int lr   = lane & 15;
        const int arow = rt * 16 + lr;  // local query row for A fragments
        const int bcol = ct * 16 + lr;  // output column for B/C/D fragments
        const bool cok = (bcol < D);

        // GEMM1: acc[16 x 4C] @ Wc[4C x D], K-steps of 4
        v8f c1 = {};
#pragma unroll 4
        for (int kk = 0; kk < C; kk++) {
            int kb = kk * 4 + half * 2;
            v2f a, b;
            a[0] = s_acc[arow * ASTR + kb + 0];
            a[1] = s_acc[arow * ASTR + kb + 1];
            b[0] = cok ? Wc[(kb + 0) * D + bcol] : 0.0f;
            b[1] = cok ? Wc[(kb + 1) * D + bcol] : 0.0f;
            c1 = __builtin_amdgcn_wmma_f32_16x16x4_f32(false, a, false, b,
                                                       (short)0, c1, false, false);
        }
        // GEMM2: h_in[16 x C] @ W[C x D]
        v8f c2 = {};
#pragma unroll 4
        for (int kk = 0; kk < C / 4; kk++) {
            int kb = kk * 4 + half * 2;
            v2f a, b;
            a[0] = s_hin[arow * HSTR + kb + 0];
            a[1] = s_hin[arow * HSTR + kb + 1];
            b[0] = cok ? W[(kb + 0) * D + bcol] : 0.0f;
            b[1] = cok ? W[(kb + 1) * D + bcol] : 0.0f;
            c2 = __builtin_amdgcn_wmma_f32_16x16x4_f32(false, a, false, b,
                                                       (short)0, c2, false, false);
        }

        const float bv = cok ? bias[bcol] : 0.0f;
        // D layout: VGPR r -> lanes0-15: M=r, lanes16-31: M=r+8; N=lr
#pragma unroll
        for (int r = 0; r < 8; r++) {
            int m    = r + half * 8;
            int grow = n0 + rt * 16 + m;
            float v1 = c1[r];
            if (RELU_CC) v1 = fmaxf(v1, 0.0f);
            float v = v1 + c2[r] + bv;
            if (grow < N && cok) out[(size_t)grow * D + bcol] = v;
        }
    }
}

// ---------------------------------------------------------------------------
extern "C" void kernel_launch(void* const* d_in, const int* in_sizes, int n_in,
                              void* d_out, int out_size, void* d_ws, size_t ws_size,
                              hipStream_t stream)
{
    (void)n_in; (void)out_size; (void)ws_size;
    const float* dy_pos   = (const float*)d_in[0];
    const float* dy_feat  = (const float*)d_in[1];
    const float* box_pos  = (const float*)d_in[2];
    const float* box_feat = (const float*)d_in[3];
    const int*   dy_idx   = (const int*)d_in[4];
    const int*   box_idx  = (const int*)d_in[5];
    const float* W1  = (const float*)d_in[6];
    const float* b1  = (const float*)d_in[7];
    const float* Wc1 = (const float*)d_in[8];
    const float* W2  = (const float*)d_in[9];
    const float* b2  = (const float*)d_in[10];
    const float* Wc2 = (const float*)d_in[11];
    const float* W3  = (const float*)d_in[12];
    const float* b3  = (const float*)d_in[13];
    const float* Wc3 = (const float*)d_in[14];
    const float* W4  = (const float*)d_in[15];
    const float* b4  = (const float*)d_in[16];
    const float* Wc4 = (const float*)d_in[17];

    const int N = in_sizes[0] / 3;

    float* h1 = (float*)d_ws;                 // [N, 96]
    float* h2 = h1 + (size_t)N * 96;          // [N, 64]
    float* h3 = h1;                           // reuse h1 region (dead after layer 2)
    float* out = (float*)d_out;               // [N, 3]

    dim3 blk(256);
    layer1_kernel<<<dim3((N + 255) / 256), blk, 0, stream>>>(
        dy_pos, dy_feat, box_pos, box_feat, dy_idx, box_idx, W1, b1, Wc1, h1, N);

    const int nblk = (N + 31) / 32;
    cconv_layer<96, 64, true ><<<dim3(nblk), blk, 0, stream>>>(
        dy_pos, dy_pos, h1, dy_idx, Wc2, W2, b2, h2, N);
    cconv_layer<64, 64, false><<<dim3(nblk), blk, 0, stream>>>(
        dy_pos, dy_pos, h2, dy_idx, Wc3, W3, b3, h3, N);
    cconv_layer<64, 3,  false><<<dim3(nblk), blk, 0, stream>>>(
        dy_pos, dy_pos, h3, dy_idx, Wc4, W4, b4, out, N);
}